// LeafMP_9225589752114
// MI455X (gfx1250) — compile-verified
//
#include <hip/hip_runtime.h>
#include <math.h>

#define DIM 128
#define GAMMA 0.5f

typedef __attribute__((ext_vector_type(2))) float v2f;
typedef __attribute__((ext_vector_type(8))) float v8f;

static __device__ __forceinline__ v8f wmma_f32_k4(v2f a, v2f b, v8f c) {
  // D = A(16x4,f32) * B(4x16,f32) + C(16x16,f32), exact fp32 matrix pipe
  return __builtin_amdgcn_wmma_f32_16x16x4_f32(false, a, false, b, (short)0, c,
                                               false, false);
}

// ---------------------------------------------------------------------------
// K0a: fuse Time2Vec + gate linear layers, emitting the TRANSPOSED matrix:
//     WfT[j][i] = (W_t2v @ W_t)[i][j],  b_f = b_t2v @ W_t + b_t
// ---------------------------------------------------------------------------
__global__ void k_fuse(const float* __restrict__ Wt2v, const float* __restrict__ Wt,
                       const float* __restrict__ bt2v, const float* __restrict__ bt,
                       float* __restrict__ WfT, float* __restrict__ bf) {
  const int i = blockIdx.x, j = threadIdx.x;
  float s = 0.0f;
  for (int k = 0; k < DIM; ++k) s = fmaf(Wt2v[i * DIM + k], Wt[k * DIM + j], s);
  WfT[j * DIM + i] = s;
  if (i == 0) {
    float sb = bt[j];
    for (int k = 0; k < DIM; ++k) sb = fmaf(bt2v[k], Wt[k * DIM + j], sb);
    bf[j] = sb;
  }
}

// K0b: transpose a 128x128 weight so B-fragments become contiguous b64 loads.
__global__ void k_transpose(const float* __restrict__ W, float* __restrict__ WT) {
  const int i = blockIdx.x, j = threadIdx.x;  // coalesced read, tiny one-off
  WT[j * DIM + i] = W[i * DIM + j];
}

// ---------------------------------------------------------------------------
// K1: per-edge precompute via WMMA. One wave (32 lanes) per 16-edge tile.
//     pe_g[e] = (H_e[e]@W_e + be_lin + b_e) * sigmoid(f(t_e)@W_f + b_f)
//     msg [e] =  H_e[e]@W_g + b_g
// A/F panels are register-resident (v2f[32]); B panels are transposed so each
// fragment is one contiguous 8-byte load.
// ---------------------------------------------------------------------------
__global__ __launch_bounds__(32) void k_edge(
    const float* __restrict__ He, const float* __restrict__ ts,
    const float* __restrict__ WeT, const float* __restrict__ beL,
    const float* __restrict__ be, const float* __restrict__ WgT,
    const float* __restrict__ bg, const float* __restrict__ omega,
    const float* __restrict__ WfT, const float* __restrict__ bf,
    float* __restrict__ pe_g, float* __restrict__ msg, int NE) {
  __shared__ float fsh[16 * DIM];  // time2vec features for the 16 edges
  const int lane = threadIdx.x;
  const int e0 = blockIdx.x << 4;

  // Build f[m][k] = k<64 ? cos(t*omega[k]) : sin(t*omega[k-64]) into LDS.
  for (int it = 0; it < (16 * DIM) / 32; ++it) {
    const int idx = it * 32 + lane;
    const int m = idx >> 7, k = idx & (DIM - 1);
    int er = e0 + m; if (er >= NE) er = NE - 1;
    const float ph = ts[er] * omega[k & 63];
    fsh[idx] = (k < 64) ? __cosf(ph) : __sinf(ph);
  }
  __syncthreads();  // single-wave workgroup: near-free

  const int m = lane & 15, kh = lane >> 4;
  int er = e0 + m; if (er >= NE) er = NE - 1;
  const float* __restrict__ Arow = He + (size_t)er * DIM;
  const float* __restrict__ Frow = fsh + m * DIM;

  // Register-resident A (H_e rows) and F (time2vec) panels: 16x128 each.
  v2f Ablk[32], Fblk[32];
#pragma unroll
  for (int kk = 0; kk < 32; ++kk) {
    const int kb = (kk << 2) + (kh << 1);  // K = 2*laneHalf + vgpr
    Ablk[kk].x = Arow[kb]; Ablk[kk].y = Arow[kb + 1];  // global b64
    Fblk[kk].x = Frow[kb]; Fblk[kk].y = Frow[kb + 1];  // ds b64
  }

#pragma unroll 1
  for (int nt = 0; nt < 8; ++nt) {
    const int col = (nt << 4) + m;  // D-matrix column = lane%16
    const float* __restrict__ Be = WeT + (size_t)col * DIM;
    const float* __restrict__ Bg = WgT + (size_t)col * DIM;
    const float* __restrict__ Bf = WfT + (size_t)col * DIM;
    v8f accE = {}, accG = {}, accF = {};
#pragma unroll
    for (int kk = 0; kk < 32; ++kk) {
      const int kb = (kk << 2) + (kh << 1);
      v2f bE; bE.x = Be[kb]; bE.y = Be[kb + 1];  // contiguous b64
      v2f bG; bG.x = Bg[kb]; bG.y = Bg[kb + 1];
      v2f bF; bF.x = Bf[kb]; bF.y = Bf[kb + 1];
      accE = wmma_f32_k4(Ablk[kk], bE, accE);
      accG = wmma_f32_k4(Ablk[kk], bG, accG);
      accF = wmma_f32_k4(Fblk[kk], bF, accF);
    }
    const float biasE = beL[col] + be[col];
    const float biasG = bg[col];
    const float biasF = bf[col];
#pragma unroll
    for (int v = 0; v < 8; ++v) {
      const int e = e0 + v + (kh << 3);  // D row = v + 8*(lane/16)
      if (e < NE) {
        const float g = 1.0f / (1.0f + __expf(-(accF[v] + biasF)));
        pe_g[(size_t)e * DIM + col] = (accE[v] + biasE) * g;
        msg[(size_t)e * DIM + col] = accG[v] + biasG;
      }
    }
  }
}

// ---------------------------------------------------------------------------
// K2: per-code projection via WMMA: proj_c = H_c @ W_c + bc_lin + b_c
// ---------------------------------------------------------------------------
__global__ __launch_bounds__(32) void k_code(
    const float* __restrict__ Hc, const float* __restrict__ WcT,
    const float* __restrict__ bcL, const float* __restrict__ bc,
    float* __restrict__ projc, int NC) {
  const int lane = threadIdx.x;
  const int c0 = blockIdx.x << 4;
  const int m = lane & 15, kh = lane >> 4;
  int cr = c0 + m; if (cr >= NC) cr = NC - 1;
  const float* __restrict__ Arow = Hc + (size_t)cr * DIM;

  v2f Ablk[32];
#pragma unroll
  for (int kk = 0; kk < 32; ++kk) {
    const int kb = (kk << 2) + (kh << 1);
    Ablk[kk].x = Arow[kb]; Ablk[kk].y = Arow[kb + 1];
  }

#pragma unroll 1
  for (int nt = 0; nt < 8; ++nt) {
    const int col = (nt << 4) + m;
    const float* __restrict__ Bc = WcT + (size_t)col * DIM;
    v8f acc = {};
#pragma unroll
    for (int kk = 0; kk < 32; ++kk) {
      const int kb = (kk << 2) + (kh << 1);
      v2f b; b.x = Bc[kb]; b.y = Bc[kb + 1];
      acc = wmma_f32_k4(Ablk[kk], b, acc);
    }
    const float bias = bcL[col] + bc[col];
#pragma unroll
    for (int v = 0; v < 8; ++v) {
      const int c = c0 + v + (kh << 3);
      if (c < NC) projc[(size_t)c * DIM + col] = acc[v] + bias;
    }
  }
}

// ---------------------------------------------------------------------------
// K3: per-incidence score = <pe_g[edge], proj_c[code]> (both L2-resident)
// ---------------------------------------------------------------------------
__global__ void k_score(const float* __restrict__ pe_g,
                        const float* __restrict__ projc,
                        const int* __restrict__ eid, const int* __restrict__ sid,
                        float* __restrict__ scores, int T) {
  const int t = blockIdx.x * blockDim.x + threadIdx.x;
  if (t >= T) return;
  const float4* __restrict__ a =
      reinterpret_cast<const float4*>(pe_g + (size_t)eid[t] * DIM);
  const float4* __restrict__ b =
      reinterpret_cast<const float4*>(projc + (size_t)sid[t] * DIM);
  float s = 0.0f;
#pragma unroll
  for (int i = 0; i < DIM / 4; ++i) {
    const float4 x = a[i], y = b[i];
    s = fmaf(x.x, y.x, s); s = fmaf(x.y, y.y, s);
    s = fmaf(x.z, y.z, s); s = fmaf(x.w, y.w, s);
  }
  scores[t] = s;
}

// ---------------------------------------------------------------------------
// K4: one 128-thread block per code; seg_ids sorted -> binary-search segment,
// block-local softmax (no global atomics), thread d owns output dim d.
// ---------------------------------------------------------------------------
__global__ __launch_bounds__(DIM) void k_agg(
    const float* __restrict__ scores, const float* __restrict__ msg,
    const int* __restrict__ eid, const int* __restrict__ sid,
    const float* __restrict__ Hc, float* __restrict__ out, int T) {
  const int c = blockIdx.x;
  const int d = threadIdx.x;

  int lo, hi;
  { int l = 0, r = T; while (l < r) { int md = (l + r) >> 1; if (sid[md] < c) l = md + 1; else r = md; } lo = l; }
  { int l = lo, r = T; while (l < r) { int md = (l + r) >> 1; if (sid[md] < c + 1) l = md + 1; else r = md; } hi = l; }

  const float hc = Hc[(size_t)c * DIM + d];
  if (hi <= lo) { out[(size_t)c * DIM + d] = hc; return; }  // uniform per block

  __shared__ float red[DIM];
  // segment max
  float mx = -INFINITY;
  for (int t = lo + d; t < hi; t += DIM) mx = fmaxf(mx, scores[t]);
  red[d] = mx; __syncthreads();
#pragma unroll
  for (int s = DIM / 2; s > 0; s >>= 1) {
    if (d < s) red[d] = fmaxf(red[d], red[d + s]);
    __syncthreads();
  }
  const float smax = red[0]; __syncthreads();
  // segment denom
  float sum = 0.0f;
  for (int t = lo + d; t < hi; t += DIM) sum += __expf(scores[t] - smax);
  red[d] = sum; __syncthreads();
#pragma unroll
  for (int s = DIM / 2; s > 0; s >>= 1) {
    if (d < s) red[d] += red[d + s];
    __syncthreads();
  }
  const float inv_denom = 1.0f / fmaxf(red[0], 1e-30f);
  // weighted aggregate: thread d owns dim d of this code's output
  float acc = 0.0f;
  for (int t = lo; t < hi; ++t) {
    const float w = __expf(scores[t] - smax) * inv_denom;
    acc = fmaf(w, msg[(size_t)eid[t] * DIM + d], acc);
  }
  out[(size_t)c * DIM + d] = GAMMA * acc + (1.0f - GAMMA) * hc;
}

// ---------------------------------------------------------------------------
extern "C" void kernel_launch(void* const* d_in, const int* in_sizes, int n_in,
                              void* d_out, int out_size, void* d_ws, size_t ws_size,
                              hipStream_t stream) {
  const float* He    = (const float*)d_in[0];
  const float* Hc    = (const float*)d_in[1];
  const float* ts    = (const float*)d_in[2];
  const float* We    = (const float*)d_in[3];
  const float* beL   = (const float*)d_in[4];
  const float* be    = (const float*)d_in[5];
  const float* Wc    = (const float*)d_in[6];
  const float* bcL   = (const float*)d_in[7];
  const float* bc    = (const float*)d_in[8];
  const float* Wt    = (const float*)d_in[9];
  const float* bt    = (const float*)d_in[10];
  const float* Wg    = (const float*)d_in[11];
  const float* bg    = (const float*)d_in[12];
  const float* omega = (const float*)d_in[13];
  const float* Wt2v  = (const float*)d_in[14];
  const float* bt2v  = (const float*)d_in[15];
  const int*   eid   = (const int*)d_in[16];
  const int*   sid   = (const int*)d_in[17];
  float* out = (float*)d_out;

  const int NE = in_sizes[2];        // timestamps: one per hyperedge
  const int NC = in_sizes[1] / DIM;  // H_c rows
  const int T  = in_sizes[16];       // incidences
  (void)n_in; (void)out_size; (void)ws_size;

  // workspace carve-out (~131.5 MB)
  char* ws = (char*)d_ws;
  size_t off = 0;
  auto take = [&](size_t bytes) {
    char* p = ws + off;
    off = (off + bytes + 255) & ~(size_t)255;
    return p;
  };
  float* WfT    = (float*)take((size_t)DIM * DIM * sizeof(float));
  float* WeT    = (float*)take((size_t)DIM * DIM * sizeof(float));
  float* WgT    = (float*)take((size_t)DIM * DIM * sizeof(float));
  float* WcT    = (float*)take((size_t)DIM * DIM * sizeof(float));
  float* bf     = (float*)take((size_t)DIM * sizeof(float));
  float* pe_g   = (float*)take((size_t)NE * DIM * sizeof(float));
  float* msgbuf = (float*)take((size_t)NE * DIM * sizeof(float));
  float* projc  = (float*)take((size_t)NC * DIM * sizeof(float));
  float* scores = (float*)take((size_t)T * sizeof(float));

  k_fuse<<<DIM, DIM, 0, stream>>>(Wt2v, Wt, bt2v, bt, WfT, bf);
  k_transpose<<<DIM, DIM, 0, stream>>>(We, WeT);
  k_transpose<<<DIM, DIM, 0, stream>>>(Wg, WgT);
  k_transpose<<<DIM, DIM, 0, stream>>>(Wc, WcT);

  const int tilesE = (NE + 15) / 16;
  k_edge<<<tilesE, 32, 0, stream>>>(He, ts, WeT, beL, be, WgT, bg, omega, WfT,
                                    bf, pe_g, msgbuf, NE);

  const int tilesC = (NC + 15) / 16;
  k_code<<<tilesC, 32, 0, stream>>>(Hc, WcT, bcL, bc, projc, NC);

  k_score<<<(T + 255) / 256, 256, 0, stream>>>(pe_g, projc, eid, sid, scores, T);

  k_agg<<<NC, DIM, 0, stream>>>(scores, msgbuf, eid, sid, Hc, out, T);
}